// SimpleGrid_86414741996199
// MI455X (gfx1250) — compile-verified
//
#include <hip/hip_runtime.h>
#include <stdint.h>

// Trilinear grid interpolation, MI455X (gfx1250).
//  - grid (64 MB) is L2-resident (192 MB L2): gathers are L2 hits -> latency-bound.
//  - coordinate stream is staged global->LDS with CDNA5 async-LDS ops (ASYNCcnt),
//    double-buffered per wave so the next tile loads while this tile gathers.
//  - output uses non-temporal stores to avoid polluting L2 (keep grid resident).
//  - WMMA intentionally unused: elementwise 8-term dot per point has no matmul
//    structure; compute is ~100 MFLOP total and is not the bottleneck.

#define BLOCK 256
#define WAVES (BLOCK / 32)

__device__ __forceinline__ void stage_point_async(float* lds_dst, const float* gsrc) {
#if defined(__gfx1250__)
  uint32_t ldsoff = (uint32_t)(uintptr_t)lds_dst;          // low 32 bits = LDS offset
  uint64_t gaddr  = (uint64_t)(uintptr_t)gsrc;
  asm volatile(
      "global_load_async_to_lds_b32 %0, %1, off\n\t"
      "global_load_async_to_lds_b32 %0, %1, off offset:4\n\t"
      "global_load_async_to_lds_b32 %0, %1, off offset:8"
      :
      : "v"(ldsoff), "v"(gaddr)
      : "memory");
#else
  lds_dst[0] = gsrc[0]; lds_dst[1] = gsrc[1]; lds_dst[2] = gsrc[2];
#endif
}

__device__ __forceinline__ void wait_async_le3() {
#if defined(__gfx1250__)
#if __has_builtin(__builtin_amdgcn_s_wait_asynccnt)
  __builtin_amdgcn_s_wait_asynccnt(3);
#else
  asm volatile("s_wait_asynccnt 3" ::: "memory");
#endif
#endif
}

__device__ __forceinline__ void wait_async_le0() {
#if defined(__gfx1250__)
#if __has_builtin(__builtin_amdgcn_s_wait_asynccnt)
  __builtin_amdgcn_s_wait_asynccnt(0);
#else
  asm volatile("s_wait_asynccnt 0" ::: "memory");
#endif
#endif
}

__global__ __launch_bounds__(BLOCK) void trilerp_gather_kernel(
    const float* __restrict__ xs,     // [N,3]
    const float* __restrict__ grid,   // [256,256,256]
    const float* __restrict__ lower,  // [3]
    const float* __restrict__ resv,   // [3]
    float* __restrict__ out,          // [N]
    int N) {
  __shared__ float sbuf[WAVES][2][96];   // per-wave double buffer: 32 pts x 3 floats

  const int lane = (int)(threadIdx.x & 31u);
  const int wave = (int)(threadIdx.x >> 5u);
  const int stride = (int)(gridDim.x * (unsigned)BLOCK);
  int base = (int)(blockIdx.x * (unsigned)BLOCK) + wave * 32;

  const float l0 = lower[0], l1 = lower[1], l2 = lower[2];
  const float ir0 = 1.0f / resv[0], ir1 = 1.0f / resv[1], ir2 = 1.0f / resv[2];
  const float bmax = 255.0f;             // size - 1

  int cur = 0;
  if (base < N) {                        // wave-uniform: lane 0 always active
    int i = base + lane;
    if (i < N) stage_point_async(&sbuf[wave][0][lane * 3], xs + (size_t)3 * i);
  }

  for (; base < N; base += stride) {
    const int nbase = base + stride;
    const bool pre = (nbase < N);        // wave-uniform
    if (pre) {                           // prefetch next tile into other buffer
      int i = nbase + lane;
      if (i < N) stage_point_async(&sbuf[wave][cur ^ 1][lane * 3], xs + (size_t)3 * i);
    }
    if (pre) wait_async_le3();           // current tile's 3 ops retired (in-order)
    else     wait_async_le0();

    const int i = base + lane;
    if (i < N) {
      const float* p = &sbuf[wave][cur][lane * 3];
      const float ix = (p[0] - l0) * ir0;
      const float iy = (p[1] - l1) * ir1;
      const float iz = (p[2] - l2) * ir2;

      const bool valid = (ix >= 0.0f) & (ix <= bmax) &
                         (iy >= 0.0f) & (iy <= bmax) &
                         (iz >= 0.0f) & (iz <= bmax);

      const float fx = floorf(ix), cx = ceilf(ix);
      const float fy = floorf(iy), cy = ceilf(iy);
      const float fz = floorf(iz), cz = ceilf(iz);

      // side distances with the reference's degenerate fix (dist==0 -> 1)
      float ax = cx - ix; ax = (ax == 0.0f) ? 1.0f : ax;
      float bx = ix - fx; bx = (bx == 0.0f) ? 1.0f : bx;
      float ay = cy - iy; ay = (ay == 0.0f) ? 1.0f : ay;
      float by = iy - fy; by = (by == 0.0f) ? 1.0f : by;
      float az = cz - iz; az = (az == 0.0f) ? 1.0f : az;
      float bz = iz - fz; bz = (bz == 0.0f) ? 1.0f : bz;

      // clamped integer corner coords (memory-safe even for invalid points)
      const int x0 = min(max((int)fx, 0), 255), x1 = min(max((int)cx, 0), 255);
      const int y0 = min(max((int)fy, 0), 255), y1 = min(max((int)cy, 0), 255);
      const int z0 = min(max((int)fz, 0), 255), z1 = min(max((int)cz, 0), 255);

      // row offsets: flat = x*65536 + y*256 + z
      const int r00 = (x0 << 16) | (y0 << 8);
      const int r01 = (x0 << 16) | (y1 << 8);
      const int r10 = (x1 << 16) | (y0 << 8);
      const int r11 = (x1 << 16) | (y1 << 8);

      // 8 gathers (L2-resident grid), folded along z first
      const float s00 = grid[r00 + z0] * az + grid[r00 + z1] * bz;
      const float s01 = grid[r01 + z0] * az + grid[r01 + z1] * bz;
      const float s10 = grid[r10 + z0] * az + grid[r10 + z1] * bz;
      const float s11 = grid[r11 + z0] * az + grid[r11 + z1] * bz;

      const float num = ay * (ax * s00 + bx * s10) + by * (ax * s01 + bx * s11);
      const float den = (ax + bx) * (ay + by) * (az + bz);

      const float r = valid ? (num / den) : 0.0f;
      __builtin_nontemporal_store(r, out + i);   // streaming store, keep L2 for grid
    }
    cur ^= 1;
  }
}

extern "C" void kernel_launch(void* const* d_in, const int* in_sizes, int n_in,
                              void* d_out, int out_size, void* d_ws, size_t ws_size,
                              hipStream_t stream) {
  const float* xs    = (const float*)d_in[0];
  const float* grid  = (const float*)d_in[1];
  const float* lower = (const float*)d_in[2];
  const float* resv  = (const float*)d_in[3];
  float* out = (float*)d_out;

  const int N = in_sizes[0] / 3;

  int blocks = 2048;                       // persistent: ~4 tiles/wave for double-buffering
  const int needed = (N + BLOCK - 1) / BLOCK;
  if (blocks > needed) blocks = needed;
  if (blocks < 1) blocks = 1;

  trilerp_gather_kernel<<<blocks, BLOCK, 0, stream>>>(xs, grid, lower, resv, out, N);
}